// InteractionModel_76897094467622
// MI455X (gfx1250) — compile-verified
//
#include <hip/hip_runtime.h>
#include <math.h>

#define A_DIM 64
#define F_DIM 12          // 11 gaussian bins + 1 scalar SH
#define G_DIM (F_DIM * A_DIM)   // 768
#define NMU 11
// tp_norm / n_pow = (1/sqrt(12*64)) / sqrt(10000)
#define MSG_SCALE 3.60843918e-04f
#define KCAL2MEV 43.3641f

typedef __attribute__((ext_vector_type(2))) float v2f;
typedef __attribute__((ext_vector_type(8))) float v8f;

__device__ __forceinline__ float silu(float x) { return x / (1.0f + expf(-x)); }

// ---------------- y = W_embed[z] + b_embed ----------------
__global__ void embed_kernel(const int* __restrict__ z, const float* __restrict__ We,
                             const float* __restrict__ be, float* __restrict__ y, int N) {
    int i = blockIdx.x * blockDim.x + threadIdx.x;
    if (i < N * A_DIM) {
        int n = i >> 6, a = i & 63;
        y[i] = We[z[n] * A_DIM + a] + be[a];
    }
}

// ---------------- edge gaussian features [E,12] ----------------
__global__ void edge_feat_kernel(const float* __restrict__ pos, const int* __restrict__ src,
                                 const int* __restrict__ dst, float* __restrict__ ef, int E) {
    int e = blockIdx.x * blockDim.x + threadIdx.x;
    if (e >= E) return;
    int s = src[e], d = dst[e];
    float dx = pos[s * 3 + 0] - pos[d * 3 + 0];
    float dy = pos[s * 3 + 1] - pos[d * 3 + 1];
    float dz = pos[s * 3 + 2] - pos[d * 3 + 2];
    float r = sqrtf(dx * dx + dy * dy + dz * dz);
#pragma unroll
    for (int j = 0; j < NMU; ++j) {
        float t = r - 0.5f * (float)j;
        ef[(size_t)e * F_DIM + j] = expf(-2.0f * t * t);   // exp(-0.5*((r-mu)/0.5)^2)
    }
    ef[(size_t)e * F_DIM + NMU] = 1.0f;                    // Y_00 scalar SH
}

// ---------------- CSR row pointers from sorted src ----------------
__global__ void row_ptr_kernel(const int* __restrict__ src, int* __restrict__ rp, int N, int E) {
    int e = blockIdx.x * blockDim.x + threadIdx.x;
    if (e > E) return;
    if (e == 0) {
        for (int n = 0; n <= src[0]; ++n) rp[n] = 0;
    } else if (e == E) {
        for (int n = src[E - 1] + 1; n <= N; ++n) rp[n] = E;
    } else {
        int a = src[e - 1], b = src[e];
        for (int n = a + 1; n <= b; ++n) rp[n] = e;
    }
}

// ---------------- g[n, f*64+b] = sum_a y[n,a] * W_tp[l,f,a,b]  (fp32 WMMA) ----------------
// GEMM [N,64] x [64,768].  Block = 64 node rows (4 row-tiles), 4 waves, each wave owns
// 12 col-tiles of 16.  Per tile: all 16 B fragments are preloaded into registers as one
// load clause (immediate offsets off one base pair), then 16 k-steps x 4 independent
// WMMAs drain while later loads are still in flight (staggered s_wait_loadcnt).
__global__ __launch_bounds__(128) void node_tp_wmma(const float* __restrict__ y,
                                                    const float* __restrict__ Wl, // [12,64,64]
                                                    float* __restrict__ g, int N) {
    __shared__ float sy[64 * A_DIM];                 // 16 KB
    int nodeBase = blockIdx.x * 64;
    bool full = (nodeBase + 64 <= N);
    // stage 64x64 y tile via float4 loads: 1024 vec4, 8 per thread
    for (int i = threadIdx.x * 4; i < 64 * A_DIM; i += 128 * 4) {
        int r = i >> 6, c = i & 63;
        int row = nodeBase + r;
        if (row >= N) row = N - 1;
        *(float4*)(sy + i) = *(const float4*)(y + (size_t)row * A_DIM + c);
    }
    __syncthreads();

    int wave = threadIdx.x >> 5;
    int lane = threadIdx.x & 31;
    int m    = lane & 15;       // A: M row / B,D: N col
    int hi   = lane >> 4;       // lane-half: K pair select / D row half

    for (int t = 0; t < 12; ++t) {
        int tile    = wave * 12 + t;          // 0..47
        int colBase = tile << 4;              // column in [0,768)
        int f       = colBase >> 6;           // 64-wide W_tp[f] block
        int bcol    = (colBase & 63) + m;
        // per-tile B base: rows (2*hi + k0), column bcol; k0 advances via immediate offsets
        const float* Bp = Wl + (size_t)f * (A_DIM * A_DIM) + 2 * hi * A_DIM + bcol;

        // preload entire tile of B fragments -> one 32-load clause, pipelined vs WMMA
        v2f bfr[16];
#pragma unroll
        for (int k = 0; k < 16; ++k) {
            bfr[k].x = Bp[k * 4 * A_DIM];
            bfr[k].y = Bp[k * 4 * A_DIM + A_DIM];
        }

        v8f c0 = {}, c1 = {}, c2 = {}, c3 = {};
#pragma unroll
        for (int k = 0; k < 16; ++k) {
            int ka = k * 4 + 2 * hi;
            v2f a0; a0.x = sy[(m     ) * A_DIM + ka]; a0.y = sy[(m     ) * A_DIM + ka + 1];
            v2f a1; a1.x = sy[(m + 16) * A_DIM + ka]; a1.y = sy[(m + 16) * A_DIM + ka + 1];
            v2f a2; a2.x = sy[(m + 32) * A_DIM + ka]; a2.y = sy[(m + 32) * A_DIM + ka + 1];
            v2f a3; a3.x = sy[(m + 48) * A_DIM + ka]; a3.y = sy[(m + 48) * A_DIM + ka + 1];
            c0 = __builtin_amdgcn_wmma_f32_16x16x4_f32(false, a0, false, bfr[k], (short)0, c0, false, false);
            c1 = __builtin_amdgcn_wmma_f32_16x16x4_f32(false, a1, false, bfr[k], (short)0, c1, false, false);
            c2 = __builtin_amdgcn_wmma_f32_16x16x4_f32(false, a2, false, bfr[k], (short)0, c2, false, false);
            c3 = __builtin_amdgcn_wmma_f32_16x16x4_f32(false, a3, false, bfr[k], (short)0, c3, false, false);
        }
        size_t colOff = (size_t)colBase + m;
        if (full) {
#pragma unroll
            for (int r = 0; r < 8; ++r) {
                int rr = hi * 8 + r;      // D: lanes0-15 M=r, lanes16-31 M=8+r
                g[(size_t)(nodeBase      + rr) * G_DIM + colOff] = c0[r];
                g[(size_t)(nodeBase + 16 + rr) * G_DIM + colOff] = c1[r];
                g[(size_t)(nodeBase + 32 + rr) * G_DIM + colOff] = c2[r];
                g[(size_t)(nodeBase + 48 + rr) * G_DIM + colOff] = c3[r];
            }
        } else {
#pragma unroll
            for (int r = 0; r < 8; ++r) {
                int rr = hi * 8 + r;
                if (nodeBase      + rr < N) g[(size_t)(nodeBase      + rr) * G_DIM + colOff] = c0[r];
                if (nodeBase + 16 + rr < N) g[(size_t)(nodeBase + 16 + rr) * G_DIM + colOff] = c1[r];
                if (nodeBase + 32 + rr < N) g[(size_t)(nodeBase + 32 + rr) * G_DIM + colOff] = c2[r];
                if (nodeBase + 48 + rr < N) g[(size_t)(nodeBase + 48 + rr) * G_DIM + colOff] = c3[r];
            }
        }
    }
}

// ---------------- per-node deterministic segment sum of SiLU messages ----------------
// block = node n (64 threads, thread t = output channel b)
__global__ __launch_bounds__(64) void interact_kernel(float* __restrict__ y,
                                                      const float* __restrict__ g,
                                                      const float* __restrict__ ef,
                                                      const int* __restrict__ rp,
                                                      const int* __restrict__ dst, int N) {
    int n = blockIdx.x;
    if (n >= N) return;
    int t = threadIdx.x;
    float acc = 0.0f;
    int e0 = rp[n], e1 = rp[n + 1];
    for (int e = e0; e < e1; ++e) {
        int d = dst[e];
        const float* ge = g + (size_t)d * G_DIM + t;          // L2-resident gather
        const float* fe = ef + (size_t)e * F_DIM;             // uniform -> scalar loads
        float msum = 0.0f;
#pragma unroll
        for (int f = 0; f < F_DIM; ++f) msum += fe[f] * ge[f * A_DIM];
        acc += silu(msum * MSG_SCALE);
    }
    y[(size_t)n * A_DIM + t] += acc;
}

// ---------------- readout: per-node silu(y@W1+b1)@W2 + b2, block-reduced ----------------
__global__ __launch_bounds__(256) void readout_kernel(const float* __restrict__ y,
                                                      const float* __restrict__ W1,
                                                      const float* __restrict__ b1,
                                                      const float* __restrict__ W2,
                                                      const float* __restrict__ b2,
                                                      float* __restrict__ partial, int N) {
    __shared__ float red[256];
    int n = blockIdx.x * 256 + threadIdx.x;
    float r = 0.0f;
    if (n < N) {
        float yr[A_DIM];
#pragma unroll
        for (int a = 0; a < A_DIM; ++a) yr[a] = y[(size_t)n * A_DIM + a];
        for (int mm = 0; mm < 128; ++mm) {
            float s = b1[mm];
#pragma unroll
            for (int a = 0; a < A_DIM; ++a) s += yr[a] * W1[a * 128 + mm]; // uniform -> SMEM
            r += silu(s) * W2[mm];
        }
        r += b2[0];
    }
    red[threadIdx.x] = r;
    __syncthreads();
    for (int off = 128; off > 0; off >>= 1) {
        if (threadIdx.x < off) red[threadIdx.x] += red[threadIdx.x + off];
        __syncthreads();
    }
    if (threadIdx.x == 0) partial[blockIdx.x] = red[0];
}

__global__ void finalize_kernel(const float* __restrict__ partial, int nPart,
                                const float* __restrict__ energy,
                                float* __restrict__ out, int N) {
    __shared__ float red[256];
    float v = 0.0f;
    for (int i = threadIdx.x; i < nPart; i += 256) v += partial[i];
    red[threadIdx.x] = v;
    __syncthreads();
    for (int off = 128; off > 0; off >>= 1) {
        if (threadIdx.x < off) red[threadIdx.x] += red[threadIdx.x + off];
        __syncthreads();
    }
    if (threadIdx.x == 0) {
        out[0] = red[0];                                              // readout
        out[1 + (size_t)N * A_DIM] = energy[0] * KCAL2MEV * (-1.0f);  // e
    }
}

__global__ void copy_y_kernel(const float* __restrict__ y, float* __restrict__ out, int N) {
    int i = blockIdx.x * blockDim.x + threadIdx.x;
    if (i < N * A_DIM) out[1 + i] = y[i];
}

extern "C" void kernel_launch(void* const* d_in, const int* in_sizes, int n_in,
                              void* d_out, int out_size, void* d_ws, size_t ws_size,
                              hipStream_t stream) {
    const int*   z      = (const int*)d_in[0];
    const float* pos    = (const float*)d_in[1];
    const float* energy = (const float*)d_in[2];
    const int*   src    = (const int*)d_in[3];
    const int*   dst    = (const int*)d_in[4];
    const float* We     = (const float*)d_in[5];
    const float* be     = (const float*)d_in[6];
    const float* Wtp    = (const float*)d_in[7];   // [3,12,64,64]
    const float* W1     = (const float*)d_in[8];
    const float* b1     = (const float*)d_in[9];
    const float* W2     = (const float*)d_in[10];
    const float* b2     = (const float*)d_in[11];
    float* out = (float*)d_out;

    const int N = in_sizes[0];
    const int E = in_sizes[3];

    // workspace layout (floats)
    float* fws = (float*)d_ws;
    size_t off = 0;
    float* ef      = fws + off; off += (size_t)E * F_DIM;     // edge features
    float* g       = fws + off; off += (size_t)N * G_DIM;     // node TP output
    float* yw      = fws + off; off += (size_t)N * A_DIM;     // node features
    float* partial = fws + off; off += 256;                   // readout partials
    int*   rp      = (int*)(fws + off);                       // CSR row ptr [N+1]

    embed_kernel<<<(N * A_DIM + 255) / 256, 256, 0, stream>>>(z, We, be, yw, N);
    edge_feat_kernel<<<(E + 255) / 256, 256, 0, stream>>>(pos, src, dst, ef, E);
    row_ptr_kernel<<<(E + 1 + 255) / 256, 256, 0, stream>>>(src, rp, N, E);

    for (int l = 0; l < 3; ++l) {
        node_tp_wmma<<<(N + 63) / 64, 128, 0, stream>>>(
            yw, Wtp + (size_t)l * F_DIM * A_DIM * A_DIM, g, N);
        interact_kernel<<<N, 64, 0, stream>>>(yw, g, ef, rp, dst, N);
    }

    int nBlk = (N + 255) / 256;
    readout_kernel<<<nBlk, 256, 0, stream>>>(yw, W1, b1, W2, b2, partial, N);
    finalize_kernel<<<1, 256, 0, stream>>>(partial, nBlk, energy, out, N);
    copy_y_kernel<<<(N * A_DIM + 255) / 256, 256, 0, stream>>>(yw, out, N);
}